// NeuralDMD_Stage2_23416161698149
// MI455X (gfx1250) — compile-verified
//
#include <hip/hip_runtime.h>
#include <hip/hip_bf16.h>
#include <stdint.h>

// Problem dims: B=32, T=512, D=4096, L=128.
// f32 WMMA (16x16x4) everywhere: matches reference f32 precision through the
// up-to-16-deep Koopman power chain where f16 would drift.

typedef __attribute__((ext_vector_type(2))) float v2f;
typedef __attribute__((ext_vector_type(8))) float v8f;
typedef __attribute__((ext_vector_type(4))) unsigned int u32x4;
typedef __attribute__((ext_vector_type(4))) int i32x4;
typedef __attribute__((ext_vector_type(8))) int i32x8;

static __device__ inline v8f wmma4(v2f a, v2f b, v8f c) {
  // (neg_a, A, neg_b, B, c_mod, C, reuse_a, reuse_b)
  return __builtin_amdgcn_wmma_f32_16x16x4_f32(false, a, false, b, (short)0, c,
                                               false, false);
}

// ---- f32 WMMA fragment helpers (ISA 7.12.2 layouts) ----------------------
static __device__ inline v2f load_a_frag(const float* A, int lda, int m0,
                                         int kb, int lane) {
  int half = lane >> 4, row = lane & 15;
  const float* p = A + (size_t)(m0 + row) * lda + kb + 2 * half;
  v2f a; a.x = p[0]; a.y = p[1];
  return a;
}
static __device__ inline v2f load_b_frag(const float* B, int ldb, int kb,
                                         int n0, int lane) {
  int half = lane >> 4, col = lane & 15;
  const float* p = B + (size_t)(kb + 2 * half) * ldb + n0 + col;
  v2f b; b.x = p[0]; b.y = p[ldb];
  return b;
}
static __device__ inline void store_c_frag(float* C, int ldc, int m0, int n0,
                                           int lane, v8f c) {
  int half = lane >> 4, col = lane & 15;
#pragma unroll
  for (int r = 0; r < 8; ++r)
    C[(size_t)(m0 + r + 8 * half) * ldc + n0 + col] = c[r];
}
static __device__ inline void store_act(float* C, int ldc, int m0, int n0,
                                        int lane, v8f c, const float* bias,
                                        bool relu) {
  int half = lane >> 4, col = lane & 15;
  int n = n0 + col;
  float bv = bias[n];
#pragma unroll
  for (int r = 0; r < 8; ++r) {
    float v = c[r] + bv;
    if (relu) v = fmaxf(v, 0.0f);
    C[(size_t)(m0 + r + 8 * half) * ldc + n] = v;
  }
}
static __device__ inline v8f gemm_tile(const float* A, int lda, const float* B,
                                       int ldb, int m0, int n0, int Kdim,
                                       int lane) {
  v8f c = {};
  for (int kb = 0; kb < Kdim; kb += 4) {
    v2f a = load_a_frag(A, lda, m0, kb, lane);
    v2f b = load_b_frag(B, ldb, kb, n0, lane);
    c = wmma4(a, b, c);
  }
  return c;
}

// ---- async global->LDS staging (ASYNCcnt path) ---------------------------
static __device__ inline void async_b128(const float* g, float* l) {
  unsigned lds = (unsigned)(uintptr_t)l;              // LDS addr = addr[31:0]
  unsigned long long ga = (unsigned long long)(uintptr_t)g;
  asm volatile("global_load_async_to_lds_b128 %0, %1, off"
               :: "v"(lds), "v"(ga) : "memory");
}
static __device__ inline void wait_async() {
  asm volatile("s_wait_asynccnt 0x0" ::: "memory");
}
static __device__ inline void async_copy_f4(const float* g, float* l,
                                            int nfloat4, int tid, int nthr) {
  for (int f = tid; f < nfloat4; f += nthr)
    async_b128(g + (size_t)f * 4, l + (size_t)f * 4);
}

// ---- Tensor Data Mover: 2-D f32 tile, global -> LDS (TENSORcnt path) -----
// D# per CDNA5 ISA: group0 {count=1, lds_addr,
// global_addr, type=2}; group1 {mask=0, data_size=4B, dims/tile/stride};
// groups 2/3 zero (tile_dim2 = 0 => 2-D).  Issue from ONE wave only
// (EXEC is ignored by TDM, so guard with a wave-uniform branch).
static __device__ inline void tdm_load_2d(const float* g, float* l,
                                          unsigned tensor_d0, unsigned tensor_d1,
                                          unsigned tile_d0, unsigned tile_d1,
                                          unsigned long long stride0) {
  unsigned long long ga = (unsigned long long)(uintptr_t)g;
  unsigned lds = (unsigned)(uintptr_t)l;
  u32x4 g0;
  g0[0] = 1u;                                            // count=1 (user D#)
  g0[1] = lds;                                           // lds_addr [63:32]
  g0[2] = (unsigned)ga;                                  // global_addr lo
  g0[3] = (unsigned)((ga >> 32) & 0x01FFFFFFull)         // global_addr hi
          | (2u << 30);                                  // type=2 ("image")
  i32x8 g1;
  g1[0] = (int)(2u << 16);                               // data_size=2 (4B)
  g1[1] = (int)((tensor_d0 & 0xFFFFu) << 16);            // tensor_dim0 lo16
  g1[2] = (int)((tensor_d0 >> 16) | ((tensor_d1 & 0xFFFFu) << 16));
  g1[3] = (int)((tensor_d1 >> 16) | ((tile_d0 & 0xFFFFu) << 16));
  g1[4] = (int)(tile_d1 & 0xFFFFu);                      // tile_dim2 = 0
  g1[5] = (int)(stride0 & 0xFFFFFFFFull);                // dim0 stride lo
  g1[6] = (int)((stride0 >> 32) & 0xFFFFull);            // dim0 stride hi
  g1[7] = 0;                                             // dim1 stride unused
  i32x4 gz = {0, 0, 0, 0};
  i32x8 gz8 = {0, 0, 0, 0, 0, 0, 0, 0};
  // 6-arg toolchain variant: (g0, g1, g2, g3, extra, cpol)
  __builtin_amdgcn_tensor_load_to_lds(g0, g1, gz, gz, gz8, 0);
}

// ===========================================================================
// Kernel 1: encoder z0 = MLP(x0).  1 WG, 16 waves = 16 (of 2x8) 16x16 tiles.
// We1..We3 staged to LDS via async-to-LDS (ASYNCcnt path).
// ===========================================================================
__global__ void __launch_bounds__(512) k_encoder(
    const float* __restrict__ x0, const float* __restrict__ We0,
    const float* __restrict__ be0, const float* __restrict__ We1,
    const float* __restrict__ be1, const float* __restrict__ We2,
    const float* __restrict__ be2, const float* __restrict__ We3,
    const float* __restrict__ be3, float* __restrict__ z0) {
  __shared__ float hA[32 * 128];
  __shared__ float hB[32 * 128];
  __shared__ float Wt[128 * 128];
  int tid = threadIdx.x, lane = tid & 31, w = tid >> 5;
  int m0 = (w >> 3) * 16, n0 = (w & 7) * 16;

  // layer 0: [32,4096] @ [4096,128] + relu (K too large to stage; stream it)
  {
    v8f c = gemm_tile(x0, 4096, We0, 128, m0, n0, 4096, lane);
    store_act(hA, 128, m0, n0, lane, c, be0, true);
  }
  __syncthreads();

  const float* Ws[3] = {We1, We2, We3};
  const float* bs[3] = {be1, be2, be3};
  float* cur = hA;
  float* nxt = hB;
#pragma unroll
  for (int i = 0; i < 3; ++i) {
    async_copy_f4(Ws[i], Wt, 4096, tid, 512);
    wait_async();
    __syncthreads();
    float* dst = (i == 2) ? z0 : nxt;
    v8f c = gemm_tile(cur, 128, Wt, 128, m0, n0, 128, lane);
    store_act(dst, 128, m0, n0, lane, c, bs[i], i != 2);
    __syncthreads();
    float* tmp = cur; cur = nxt; nxt = tmp;
  }
}

// ===========================================================================
// Kernel 2: bit powers P[b] = K^(2^b), b=0..15.  Sequential squaring, operand
// kept in LDS (A and B fragments both read the same 64 KB tile).
// ===========================================================================
__global__ void __launch_bounds__(512) k_powers(const float* __restrict__ Kg,
                                                float* __restrict__ P) {
  __shared__ float Pa[128 * 128];
  __shared__ float Pb[128 * 128];
  int tid = threadIdx.x, lane = tid & 31, w = tid >> 5;

  const float4* K4 = (const float4*)Kg;
  float4* Pa4 = (float4*)Pa;
  float4* Pb4 = (float4*)Pb;
  float4* Pg4 = (float4*)P;
  for (int f = tid; f < 4096; f += 512) {   // P[0] = K, and stage into LDS
    float4 v = K4[f];
    Pa4[f] = v;
    Pg4[f] = v;
  }
  __syncthreads();

  for (int b = 1; b < 16; ++b) {
    float* Pout = P + (size_t)b * 16384;
#pragma unroll
    for (int it = 0; it < 4; ++it) {        // 64 tiles / 16 waves
      int t = w + 16 * it;
      int m0 = (t >> 3) * 16, n0 = (t & 7) * 16;
      v8f c = gemm_tile(Pa, 128, Pa, 128, m0, n0, 128, lane);
      store_c_frag(Pb, 128, m0, n0, lane, c);
      store_c_frag(Pout, 128, m0, n0, lane, c);
    }
    __syncthreads();
    for (int f = tid; f < 4096; f += 512) Pa4[f] = Pb4[f];
    __syncthreads();
  }
}

// ===========================================================================
// Kernel 3: Z[t] = K^{e_t} @ z0^T  via set-bit chain Y <- P_b @ Y.
// One WG per t; Y [128,32] ping-pongs in LDS; P_b tiles fetched by TDM.
// ===========================================================================
__global__ void __launch_bounds__(512) k_apply(const float* __restrict__ z0,
                                               const long long* __restrict__ idx,
                                               const float* __restrict__ P,
                                               float* __restrict__ Z) {
  __shared__ float Ya[128 * 32];
  __shared__ float Yb[128 * 32];
  __shared__ float Pt[128 * 128];
  int tid = threadIdx.x, lane = tid & 31, w = tid >> 5;
  int t = blockIdx.x;
  int e = (int)(idx[t] - idx[0]);

  for (int s = tid; s < 4096; s += 512) {   // Y = z0^T  ([128,32])
    int i = s >> 5, bc = s & 31;
    Ya[s] = z0[(size_t)bc * 128 + i];
  }
  __syncthreads();

  float* cur = Ya;
  float* nxt = Yb;
  for (int b = 0; b < 16; ++b) {
    if ((e >> b) & 1) {
      if (w == 0) {                          // single TDM DMA for the 64KB tile
        tdm_load_2d(P + (size_t)b * 16384, Pt, 128u, 128u, 128u, 128u, 128ull);
        __builtin_amdgcn_s_wait_tensorcnt((short)0);
      }
      __syncthreads();
      int m0 = (w >> 1) * 16, n0 = (w & 1) * 16;  // 8x2 tiles, 1 per wave
      v8f c = gemm_tile(Pt, 128, cur, 32, m0, n0, 128, lane);
      store_c_frag(nxt, 32, m0, n0, lane, c);
      __syncthreads();
      float* tmp = cur; cur = nxt; nxt = tmp;
    }
  }

  // Z[t, bcol, i] = Y[i, bcol]
  for (int s = tid; s < 4096; s += 512) {
    int i = s >> 5, bc = s & 31;
    Z[((size_t)t * 32 + bc) * 128 + i] = cur[s];
  }
}

// ===========================================================================
// Kernel 4: fused decoder. 256 WGs x 64 rows. All staging via TDM with
// TENSORcnt double buffering (issue next tile, s_wait_tensorcnt 1 => current
// tile landed since same-wave TDM ops retire in order).
// ===========================================================================
__global__ void __launch_bounds__(512) k_decoder(
    const float* __restrict__ Z, const float* __restrict__ Wd0,
    const float* __restrict__ bd0, const float* __restrict__ Wd1,
    const float* __restrict__ bd1, const float* __restrict__ Wd2,
    const float* __restrict__ bd2, const float* __restrict__ Wd3,
    const float* __restrict__ bd3, float* __restrict__ out) {
  __shared__ float hA[64 * 128];
  __shared__ float hB[64 * 128];
  __shared__ float WtA[128 * 128];
  __shared__ float WtB[128 * 128];
  int tid = threadIdx.x, lane = tid & 31, w = tid >> 5;
  int rb = blockIdx.x;  // row block: rows [rb*64, rb*64+64)

  float* curW = WtA;
  float* nxtW = WtB;
  if (w == 0) {  // prologue: Z row-block + first hidden weights
    tdm_load_2d(Z + (size_t)rb * 8192, hA, 128u, 16384u, 128u, 64u, 128ull);
    tdm_load_2d(Wd0, curW, 128u, 128u, 128u, 128u, 128ull);
  }

  const float* Ws[3] = {Wd0, Wd1, Wd2};
  const float* bs[3] = {bd0, bd1, bd2};
  float* cur = hA;
  float* nxt = hB;
#pragma unroll
  for (int i = 0; i < 3; ++i) {
    if (w == 0) {
      if (i < 2) tdm_load_2d(Ws[i + 1], nxtW, 128u, 128u, 128u, 128u, 128ull);
      else       tdm_load_2d(Wd3, nxtW, 4096u, 128u, 128u, 128u, 4096ull);
      __builtin_amdgcn_s_wait_tensorcnt((short)1);  // current inputs landed
    }
    __syncthreads();
#pragma unroll
    for (int it = 0; it < 2; ++it) {        // 4x8 = 32 tiles / 16 waves
      int tI = w + 16 * it;
      int m0 = (tI >> 3) * 16, n0 = (tI & 7) * 16;
      v8f c = gemm_tile(cur, 128, curW, 128, m0, n0, 128, lane);
      store_act(nxt, 128, m0, n0, lane, c, bs[i], true);
    }
    __syncthreads();
    float* tmp = cur; cur = nxt; nxt = tmp;
    float* tmpw = curW; curW = nxtW; nxtW = tmpw;
  }

  // final layer: [64,128] @ [128,4096], streamed in 32 chunks of 128 cols.
  // curW holds chunk 0 (in flight); double-buffer via TENSORcnt.
  for (int nc = 0; nc < 32; ++nc) {
    if (w == 0) {
      if (nc < 31) {
        tdm_load_2d(Wd3 + (size_t)(nc + 1) * 128, nxtW, 4096u, 128u, 128u,
                    128u, 4096ull);
        __builtin_amdgcn_s_wait_tensorcnt((short)1);   // chunk nc landed
      } else {
        __builtin_amdgcn_s_wait_tensorcnt((short)0);
      }
    }
    if (nc + 1 < 32 && tid < 128)           // warm L2 for the next chunk
      __builtin_prefetch(Wd3 + (size_t)tid * 4096 + (nc + 1) * 128, 0, 1);
    __syncthreads();
    float* Obase = out + (size_t)rb * 64 * 4096 + (size_t)nc * 128;
#pragma unroll
    for (int it = 0; it < 2; ++it) {
      int tI = w + 16 * it;
      int m0 = (tI >> 3) * 16, n0 = (tI & 7) * 16;
      v8f c = gemm_tile(cur, 128, curW, 128, m0, n0, 128, lane);
      store_act(Obase, 4096, m0, n0, lane, c, bd3 + nc * 128, false);
    }
    __syncthreads();                        // buffer reused next chunk
    float* tmpw = curW; curW = nxtW; nxtW = tmpw;
  }
}

// ===========================================================================
extern "C" void kernel_launch(void* const* d_in, const int* in_sizes, int n_in,
                              void* d_out, int out_size, void* d_ws,
                              size_t ws_size, hipStream_t stream) {
  const float* x0      = (const float*)d_in[0];
  const long long* idx = (const long long*)d_in[1];
  const float* Kop     = (const float*)d_in[2];
  const float* We0 = (const float*)d_in[3];  const float* be0 = (const float*)d_in[4];
  const float* We1 = (const float*)d_in[5];  const float* be1 = (const float*)d_in[6];
  const float* We2 = (const float*)d_in[7];  const float* be2 = (const float*)d_in[8];
  const float* We3 = (const float*)d_in[9];  const float* be3 = (const float*)d_in[10];
  const float* Wd0 = (const float*)d_in[11]; const float* bd0 = (const float*)d_in[12];
  const float* Wd1 = (const float*)d_in[13]; const float* bd1 = (const float*)d_in[14];
  const float* Wd2 = (const float*)d_in[15]; const float* bd2 = (const float*)d_in[16];
  const float* Wd3 = (const float*)d_in[17]; const float* bd3 = (const float*)d_in[18];
  float* out = (float*)d_out;

  char* ws = (char*)d_ws;
  float* z0 = (float*)(ws);                         // 32*128*4      = 16 KB
  float* P  = (float*)(ws + 16384);                 // 16*128*128*4  = 1 MB
  float* Z  = (float*)(ws + 16384 + 16 * 128 * 128 * 4);  // 16384*128*4 = 8 MB

  k_encoder<<<1, 512, 0, stream>>>(x0, We0, be0, We1, be1, We2, be2, We3, be3, z0);
  k_powers<<<1, 512, 0, stream>>>(Kop, P);
  k_apply<<<512, 512, 0, stream>>>(z0, idx, P, Z);
  k_decoder<<<256, 512, 0, stream>>>(Z, Wd0, bd0, Wd1, bd1, Wd2, bd2, Wd3, bd3, out);
}